// SCTConv_85426899517603
// MI455X (gfx1250) — compile-verified
//
#include <hip/hip_runtime.h>
#include <hip/hip_bf16.h>

#define HID 128

typedef __attribute__((ext_vector_type(2))) float v2f;
typedef __attribute__((ext_vector_type(8))) float v8f;

// ---------------------------------------------------------------------------
// SpMM: y[rows[e], :] += vals[e] * x[cols[e], :]   (y must be pre-zeroed)
// 32 threads per edge, 4 columns per thread (float4 gather, 4 f32 atomics).
// ---------------------------------------------------------------------------
__global__ __launch_bounds__(256) void spmm_kernel(
    const int* __restrict__ rows, const int* __restrict__ cols,
    const float* __restrict__ vals, const float* __restrict__ x,
    float* __restrict__ y, int nedges) {
  long long t = (long long)blockIdx.x * blockDim.x + threadIdx.x;
  int edge = (int)(t >> 5);
  if (edge >= nedges) return;
  int c = ((int)t & 31) * 4;
  int r = rows[edge];
  int s = cols[edge];
  float v = vals[edge];
  const float4 xv = *(const float4*)(x + (size_t)s * HID + c);
  float* yp = y + (size_t)r * HID + c;
  atomicAdd(yp + 0, v * xv.x);
  atomicAdd(yp + 1, v * xv.y);
  atomicAdd(yp + 2, v * xv.z);
  atomicAdd(yp + 3, v * xv.w);
}

// ---------------------------------------------------------------------------
// Elementwise absolute value over a flat range (for the P-branch outputs).
// ---------------------------------------------------------------------------
__global__ __launch_bounds__(256) void abs_kernel(float* __restrict__ p, long long n) {
  long long t = (long long)blockIdx.x * blockDim.x + threadIdx.x;
  if (t < n) p[t] = fabsf(p[t]);
}

// ---------------------------------------------------------------------------
// Per-node attention + the strided branch remix.
// One 128-thread block per node.
// ---------------------------------------------------------------------------
__device__ __forceinline__ float block_reduce128(float v, float* red) {
  int j = threadIdx.x;
  red[j] = v;
  __syncthreads();
  #pragma unroll
  for (int s = 64; s > 0; s >>= 1) {
    if (j < s) red[j] += red[j + s];
    __syncthreads();
  }
  float r = red[0];
  __syncthreads();
  return r;
}

__global__ __launch_bounds__(128) void attn_kernel(
    const float* __restrict__ X, const float* __restrict__ a,
    const float* __restrict__ hA, const float* __restrict__ hA2,
    const float* __restrict__ hA3, const float* __restrict__ hs1,
    const float* __restrict__ hs2, const float* __restrict__ hs3,
    float* __restrict__ hp) {
  __shared__ float sh[6][HID];
  __shared__ float red[HID];

  int n = blockIdx.x;
  int j = threadIdx.x;
  size_t base = (size_t)n * HID + j;

  const float* br[6] = {hA, hA2, hA3, hs1, hs2, hs3};

  float hv = X[base];
  float a_lo = a[j];
  float a_hi = a[HID + j];

  float p[6];
  #pragma unroll
  for (int k = 0; k < 6; ++k) {
    float v = br[k][base];
    sh[k][j] = v;
    float rv = v > 0.f ? v : 0.f;
    p[k] = rv * a_hi;
  }
  __syncthreads();

  float rh = hv > 0.f ? hv : 0.f;
  float ec = block_reduce128(rh * a_lo, red);

  float e6[6];
  #pragma unroll
  for (int k = 0; k < 6; ++k) e6[k] = ec + block_reduce128(p[k], red);

  // softmax over 6 (computed redundantly in every thread; values are uniform)
  float mx = e6[0];
  #pragma unroll
  for (int k = 1; k < 6; ++k) mx = fmaxf(mx, e6[k]);
  float den = 0.f;
  float att[6];
  #pragma unroll
  for (int k = 0; k < 6; ++k) { att[k] = __expf(e6[k] - mx); den += att[k]; }
  float inv = 1.0f / den;
  #pragma unroll
  for (int k = 0; k < 6; ++k) att[k] *= inv;

  // h_prime[n, j] = (1/6) * sum_k att[k] * branch[(k*128+j)%6][n, (k*128+j)/6]
  float acc = 0.f;
  #pragma unroll
  for (int k = 0; k < 6; ++k) {
    int idx = k * HID + j;
    acc += att[k] * sh[idx % 6][idx / 6];
  }
  hp[base] = acc * (1.0f / 6.0f);
}

// ---------------------------------------------------------------------------
// Dense layer: out = leaky_relu(inp @ W^T + bias), inp [nrows x 128],
// W [128 x 128] (row = output feature). Full-precision f32 WMMA:
// V_WMMA_F32_16X16X4_F32, one wave per 16x16 output tile, 32 K-steps.
// A fragment: lane L%16 = row M; lanes<16 hold K=0,1; lanes>=16 hold K=2,3.
// B fragment (B[k][n] = W[n][k]): lane L%16 = col N; same K split.
// C/D: VGPR r = row M=r (lanes 0-15) / M=8+r (lanes 16-31), N = lane%16.
// ---------------------------------------------------------------------------
__global__ __launch_bounds__(256) void dense_wmma_kernel(
    const float* __restrict__ inp, const float* __restrict__ W,
    const float* __restrict__ bias, float* __restrict__ out, int nrows) {
  int w    = threadIdx.x >> 5;
  int lane = threadIdx.x & 31;
  int lm   = lane & 15;
  int hi   = (lane >> 4) & 1;

  int m0   = blockIdx.x * 16;
  int col0 = w * 16;

  int arowi = m0 + lm;
  if (arowi >= nrows) arowi = nrows - 1;  // clamp loads; stores guarded below

  const float2* ap = (const float2*)(inp + (size_t)arowi * HID + 2 * hi);
  const float2* bp = (const float2*)(W + (size_t)(col0 + lm) * HID + 2 * hi);

  v8f c = {};
  #pragma unroll 4
  for (int k = 0; k < HID / 4; ++k) {
    float2 af = ap[2 * k];
    float2 bf = bp[2 * k];
    v2f av; av.x = af.x; av.y = af.y;
    v2f bv; bv.x = bf.x; bv.y = bf.y;
    c = __builtin_amdgcn_wmma_f32_16x16x4_f32(
        /*neg_a=*/false, av, /*neg_b=*/false, bv,
        /*c_mod=*/(short)0, c, /*reuse_a=*/false, /*reuse_b=*/false);
  }

  int col   = col0 + lm;
  int rbase = m0 + 8 * hi;
  float b   = bias[col];
  #pragma unroll
  for (int r = 0; r < 8; ++r) {
    int row = rbase + r;
    if (row < nrows) {
      float v = c[r] + b;
      out[(size_t)row * HID + col] = v > 0.f ? v : 0.01f * v;
    }
  }
}

// ---------------------------------------------------------------------------
// Launch
// ---------------------------------------------------------------------------
extern "C" void kernel_launch(void* const* d_in, const int* in_sizes, int n_in,
                              void* d_out, int out_size, void* d_ws, size_t ws_size,
                              hipStream_t stream) {
  const float* X  = (const float*)d_in[0];
  const float* a  = (const float*)d_in[1];
  const float* W1 = (const float*)d_in[2];
  const float* b1 = (const float*)d_in[3];
  const float* W2 = (const float*)d_in[4];
  const float* b2 = (const float*)d_in[5];

  const int*   Ar = (const int*)d_in[6];
  const int*   Ac = (const int*)d_in[7];
  const float* Av = (const float*)d_in[8];
  const int*   P1r = (const int*)d_in[9];
  const int*   P1c = (const int*)d_in[10];
  const float* P1v = (const float*)d_in[11];
  const int*   P2r = (const int*)d_in[12];
  const int*   P2c = (const int*)d_in[13];
  const float* P2v = (const float*)d_in[14];
  const int*   P3r = (const int*)d_in[15];
  const int*   P3c = (const int*)d_in[16];
  const float* P3v = (const float*)d_in[17];
  // d_in[18..20] = Psct (unused by reference, withgres=False)

  int N  = in_sizes[0] / HID;
  int EA = in_sizes[6];
  int E1 = in_sizes[9];
  int E2 = in_sizes[12];
  int E3 = in_sizes[15];

  size_t NH = (size_t)N * HID;
  float* ws  = (float*)d_ws;
  float* hA  = ws + 0 * NH;
  float* hA2 = ws + 1 * NH;
  float* hA3 = ws + 2 * NH;
  float* hs1 = ws + 3 * NH;
  float* hs2 = ws + 4 * NH;
  float* hs3 = ws + 5 * NH;
  float* hp  = ws + 6 * NH;
  float* mid = ws + 7 * NH;

  // Zero the six SpMM accumulators (graph-capture-safe memset node).
  hipMemsetAsync(hA, 0, 6 * NH * sizeof(float), stream);

  auto spblocks = [](int e) { return (int)(((long long)e * 32 + 255) / 256); };

  // Chained A-powers (stream order enforces the dependency).
  spmm_kernel<<<spblocks(EA), 256, 0, stream>>>(Ar, Ac, Av, X,   hA,  EA);
  spmm_kernel<<<spblocks(EA), 256, 0, stream>>>(Ar, Ac, Av, hA,  hA2, EA);
  spmm_kernel<<<spblocks(EA), 256, 0, stream>>>(Ar, Ac, Av, hA2, hA3, EA);

  // Independent scattering operators.
  spmm_kernel<<<spblocks(E1), 256, 0, stream>>>(P1r, P1c, P1v, X, hs1, E1);
  spmm_kernel<<<spblocks(E2), 256, 0, stream>>>(P2r, P2c, P2v, X, hs2, E2);
  spmm_kernel<<<spblocks(E3), 256, 0, stream>>>(P3r, P3c, P3v, X, hs3, E3);

  // |.| over hs1..hs3 (contiguous).
  long long nabs = 3 * (long long)NH;
  abs_kernel<<<(int)((nabs + 255) / 256), 256, 0, stream>>>(hs1, nabs);

  // Attention + branch remix -> h_prime.
  attn_kernel<<<N, HID, 0, stream>>>(X, a, hA, hA2, hA3, hs1, hs2, hs3, hp);

  // Two dense layers on f32 WMMA.
  int gtiles = (N + 15) / 16;
  dense_wmma_kernel<<<gtiles, 256, 0, stream>>>(hp,  W1, b1, mid, N);
  dense_wmma_kernel<<<gtiles, 256, 0, stream>>>(mid, W2, b2, (float*)d_out, N);
}